// PPNetLoss_89335319757025
// MI455X (gfx1250) — compile-verified
//
#include <hip/hip_runtime.h>
#include <math.h>

#define BATCH        4096
#define K_CONCEPTS   312
#define P_PROTOS     10
#define N_GROUPS     624            // 2*K
#define ROW_LEN      6240           // N_GROUPS * P_PROTOS
#define N_CLASSES    200
#define N_PROTOTYPES 6240

typedef __attribute__((ext_vector_type(2))) float v2f;
typedef __attribute__((ext_vector_type(8))) float v8f;

// Workspace (float) layout:
// [0] cluster_sum  [1] sep_sum  [2] target_loss_sum  [3] l1_sum
// [4 .. 4+311] per-concept BCE column sums
#define WS_CLUST 0
#define WS_SEP   1
#define WS_TGT   2
#define WS_L1    3
#define WS_CONC  4
#define WS_N     (4 + K_CONCEPTS)

__global__ void init_ws_kernel(float* __restrict__ ws) {
    int i = blockIdx.x * blockDim.x + threadIdx.x;
    if (i < WS_N) ws[i] = 0.0f;
}

__device__ __forceinline__ float bce_logits(float x, float y) {
    // max(x,0) - x*y + log1p(exp(-|x|))
    return fmaxf(x, 0.0f) - x * y + log1pf(expf(-fabsf(x)));
}

// ---------------------------------------------------------------------------
// BCE + per-concept column sums via WMMA (D = ones(16x4) * B(4x16) + C).
// Each wave owns one 16-concept tile; each WMMA folds 4 batch rows into the
// f32 accumulator. With A == all-ones the result is the column sum of B over
// K regardless of the exact K-slot lane permutation, and all 16 rows of D are
// identical, so we read element 0 (M=0, N=lane) on lanes 0..15.
// Tail tile (concepts 312..319) is handled branch-free: clamp the load index
// and cndmask the BCE value to zero, keeping EXEC all-ones for the WMMA.
// ---------------------------------------------------------------------------
__global__ __launch_bounds__(640) void bce_colsum_kernel(
        const float* __restrict__ logits,
        const int*   __restrict__ labels,
        float*       __restrict__ ws,
        int rows_per_block) {
    const int lane    = threadIdx.x & 31;
    const int wave    = threadIdx.x >> 5;          // 0..19 -> concept tile
    const int concept = wave * 16 + (lane & 15);
    const int kb      = (lane < 16) ? 0 : 2;       // which pair of batch rows
    const bool valid  = (concept < K_CONCEPTS);
    const int  cidx   = valid ? concept : 0;       // clamped, always in-range
    const int  b0     = blockIdx.x * rows_per_block;

    v2f a; a[0] = 1.0f; a[1] = 1.0f;               // ones A-matrix (16x4)
    v8f c = {};                                    // f32 accumulator (16x16)

    for (int b = b0; b < b0 + rows_per_block; b += 4) {
        const size_t i0 = (size_t)(b + kb) * K_CONCEPTS + cidx;
        const size_t i1 = i0 + K_CONCEPTS;
        const float x0 = logits[i0];
        const float x1 = logits[i1];
        const float y0 = (float)labels[i0];
        const float y1 = (float)labels[i1];
        float e0 = bce_logits(x0, y0);
        float e1 = bce_logits(x1, y1);
        e0 = valid ? e0 : 0.0f;
        e1 = valid ? e1 : 0.0f;
        v2f bm; bm[0] = e0; bm[1] = e1;
        // EXEC is all-ones here (loop body is branch-free).
        c = __builtin_amdgcn_wmma_f32_16x16x4_f32(
                /*neg_a=*/false, a, /*neg_b=*/false, bm,
                /*c_mod=*/(short)0, c, /*reuse_a=*/false, /*reuse_b=*/false);
    }

    if (lane < 16 && valid) {
        atomicAdd(&ws[WS_CONC + concept], c[0]);   // D row M=0, N=lane
    }
}

// ---------------------------------------------------------------------------
// min_distances: stage one 24.96KB row into LDS with CDNA5 async copies
// (global_load_async_to_lds_b128, tracked by ASYNCcnt, no VGPR round-trip),
// then compute the 624 group-of-10 mins and route by label bit into
// cluster/separation partial sums.
// ---------------------------------------------------------------------------
__global__ __launch_bounds__(256) void mindist_kernel(
        const float* __restrict__ md,
        const int*   __restrict__ ctrue,
        float*       __restrict__ ws) {
    __shared__ __align__(16) float rowbuf[ROW_LEN];   // 24960 B, 16B aligned
    __shared__ float gmin[N_GROUPS];
    __shared__ float red[16];

    const int b = blockIdx.x;
    const int t = threadIdx.x;

    // Async copy: 1560 x b128 transfers, global memory -> LDS directly.
    // LDS byte address = low 32 bits of the generic pointer (wg-relative).
    const unsigned lds_base = (unsigned)(uintptr_t)&rowbuf[0];
    const float*   src      = md + (size_t)b * ROW_LEN;
    for (int i = t; i < ROW_LEN / 4; i += 256) {
        const unsigned           ldsa = lds_base + (unsigned)i * 16u;
        const unsigned long long ga   = (unsigned long long)(src + i * 4);
        asm volatile("global_load_async_to_lds_b128 %0, %1, off"
                     :: "v"(ldsa), "v"(ga) : "memory");
    }
    asm volatile("s_wait_asynccnt 0x0" ::: "memory");
    __syncthreads();

    for (int g = t; g < N_GROUPS; g += 256) {
        float m = rowbuf[g * P_PROTOS];
#pragma unroll
        for (int j = 1; j < P_PROTOS; ++j)
            m = fminf(m, rowbuf[g * P_PROTOS + j]);
        gmin[g] = m;
    }
    __syncthreads();

    float cl = 0.0f, sp = 0.0f;
    for (int k = t; k < K_CONCEPTS; k += 256) {
        const int   cb = ctrue[(size_t)b * K_CONCEPTS + k];
        const float g0 = gmin[k];
        const float g1 = gmin[k + K_CONCEPTS];
        cl += cb ? g0 : g1;
        sp += cb ? g1 : g0;
    }
    for (int off = 16; off; off >>= 1) {
        cl += __shfl_xor(cl, off, 32);
        sp += __shfl_xor(sp, off, 32);
    }
    const int lane = t & 31, w = t >> 5;
    if (lane == 0) { red[w] = cl; red[8 + w] = sp; }
    __syncthreads();
    if (t == 0) {
        float c2 = 0.0f, s2 = 0.0f;
        for (int i = 0; i < 8; ++i) { c2 += red[i]; s2 += red[8 + i]; }
        atomicAdd(&ws[WS_CLUST], c2);
        atomicAdd(&ws[WS_SEP],   s2);
    }
}

// ---------------------------------------------------------------------------
// Target cross-entropy: one wave32 per batch row (max, sum-exp, gather).
// ---------------------------------------------------------------------------
__global__ __launch_bounds__(256) void target_ce_kernel(
        const float* __restrict__ logits,
        const int*   __restrict__ tgt,
        float*       __restrict__ ws) {
    const int gid  = blockIdx.x * blockDim.x + threadIdx.x;
    const int row  = gid >> 5;
    const int lane = gid & 31;
    if (row >= BATCH) return;

    const float* rp = logits + (size_t)row * N_CLASSES;
    float mx = -3.4e38f;
    for (int i = lane; i < N_CLASSES; i += 32) mx = fmaxf(mx, rp[i]);
    for (int off = 16; off; off >>= 1) mx = fmaxf(mx, __shfl_xor(mx, off, 32));

    float s = 0.0f;
    for (int i = lane; i < N_CLASSES; i += 32) s += expf(rp[i] - mx);
    for (int off = 16; off; off >>= 1) s += __shfl_xor(s, off, 32);

    if (lane == 0) {
        const float loss = logf(s) + mx - rp[tgt[row]];
        atomicAdd(&ws[WS_TGT], loss);
    }
}

// ---------------------------------------------------------------------------
// L1: sum |w * mask| over 200*6240, grid-stride + hierarchical reduction.
// ---------------------------------------------------------------------------
__global__ __launch_bounds__(256) void l1_kernel(
        const float* __restrict__ w,
        const float* __restrict__ m,
        float*       __restrict__ ws, int n) {
    float s = 0.0f;
    const int stride = gridDim.x * blockDim.x;
    for (int i = blockIdx.x * blockDim.x + threadIdx.x; i < n; i += stride)
        s += fabsf(w[i] * m[i]);
    for (int off = 16; off; off >>= 1) s += __shfl_xor(s, off, 32);
    __shared__ float red[8];
    const int lane = threadIdx.x & 31, wv = threadIdx.x >> 5;
    if (lane == 0) red[wv] = s;
    __syncthreads();
    if (threadIdx.x == 0) {
        float t = 0.0f;
        for (int i = 0; i < 8; ++i) t += red[i];
        atomicAdd(&ws[WS_L1], t);
    }
}

// ---------------------------------------------------------------------------
// Finalize: combine accumulators into the 319-float output tuple.
// out[0]=target_loss, out[1..312]=concepts_loss, out[313]=summed,
// out[314]=total, out[315]=cross_entropy, out[316]=cluster, out[317]=sep,
// out[318]=l1
// ---------------------------------------------------------------------------
__global__ void finalize_kernel(const float* __restrict__ ws,
                                float* __restrict__ out) {
    const int t = threadIdx.x;
    const float invB = 1.0f / (float)BATCH;
    for (int k = t; k < K_CONCEPTS; k += blockDim.x)
        out[1 + k] = ws[WS_CONC + k] * invB;
    if (t == 0) {
        float tot = 0.0f;
        for (int k = 0; k < K_CONCEPTS; ++k) tot += ws[WS_CONC + k];
        const float denom = (float)BATCH * (float)K_CONCEPTS;
        const float ce = tot / denom;
        const float cl = ws[WS_CLUST] / denom;
        const float sp = ws[WS_SEP]   / denom;
        const float l1 = ws[WS_L1];
        const float tl = ws[WS_TGT] * invB;
        const float summed = 1.0f * ce + 0.8f * cl + (-0.08f) * sp + 1.0e-4f * l1;
        out[0]   = tl;
        out[313] = summed;
        out[314] = tl + 1.0f * summed;   // ALPHA = 1.0
        out[315] = ce;
        out[316] = cl;
        out[317] = sp;
        out[318] = l1;
    }
}

extern "C" void kernel_launch(void* const* d_in, const int* in_sizes, int n_in,
                              void* d_out, int out_size, void* d_ws, size_t ws_size,
                              hipStream_t stream) {
    const float* concepts_pred = (const float*)d_in[0];
    const int*   concepts_true = (const int*)  d_in[1];
    const float* min_dist      = (const float*)d_in[2];
    const float* tgt_logits    = (const float*)d_in[3];
    const int*   tgt_true      = (const int*)  d_in[4];
    const float* llw           = (const float*)d_in[5];
    const float* l1m           = (const float*)d_in[6];
    float* out = (float*)d_out;
    float* ws  = (float*)d_ws;

    init_ws_kernel<<<1, 512, 0, stream>>>(ws);
    bce_colsum_kernel<<<64, 640, 0, stream>>>(concepts_pred, concepts_true, ws,
                                              BATCH / 64);
    mindist_kernel<<<BATCH, 256, 0, stream>>>(min_dist, concepts_true, ws);
    target_ce_kernel<<<(BATCH * 32) / 256, 256, 0, stream>>>(tgt_logits,
                                                             tgt_true, ws);
    l1_kernel<<<512, 256, 0, stream>>>(llw, l1m, ws, N_CLASSES * N_PROTOTYPES);
    finalize_kernel<<<1, 512, 0, stream>>>(ws, out);
}